// TransducerLoss_38414187495872
// MI455X (gfx1250) — compile-verified
//
#include <hip/hip_runtime.h>
#include <math.h>

#define B_ 8
#define T_ 128
#define U_ 64
#define V_ 512
#define NEG_INF (-1e30f)

typedef float v2f __attribute__((ext_vector_type(2)));
typedef float v8f __attribute__((ext_vector_type(8)));

// ---------------------------------------------------------------------------
// Kernel 1: per-row max + exp(x - max).  Rows = B*T emission rows then B*U
// prediction rows, each of length V=512.  128 threads/row, 4 elems/thread.
// ---------------------------------------------------------------------------
__global__ void rnnt_prep(const float* __restrict__ em, const float* __restrict__ pr,
                          float* __restrict__ Ee, float* __restrict__ Pe,
                          float* __restrict__ mE, float* __restrict__ mP) {
    __shared__ float sred[4];
    int row = blockIdx.x;
    const float* src;
    float* dst;
    float* mdst;
    if (row < B_ * T_) {
        src = em + (size_t)row * V_;
        dst = Ee + (size_t)row * V_;
        mdst = mE + row;
    } else {
        int r2 = row - B_ * T_;
        src = pr + (size_t)r2 * V_;
        dst = Pe + (size_t)r2 * V_;
        mdst = mP + r2;
    }
    int tid = threadIdx.x;
    float x0 = src[tid];
    float x1 = src[tid + 128];
    float x2 = src[tid + 256];
    float x3 = src[tid + 384];
    float m = fmaxf(fmaxf(x0, x1), fmaxf(x2, x3));
    // wave32 reduction
    #pragma unroll
    for (int off = 16; off > 0; off >>= 1)
        m = fmaxf(m, __shfl_xor(m, off, 32));
    if ((tid & 31) == 0) sred[tid >> 5] = m;
    __syncthreads();
    m = fmaxf(fmaxf(sred[0], sred[1]), fmaxf(sred[2], sred[3]));
    dst[tid]       = expf(x0 - m);
    dst[tid + 128] = expf(x1 - m);
    dst[tid + 256] = expf(x2 - m);
    dst[tid + 384] = expf(x3 - m);
    if (tid == 0) *mdst = m;
}

// ---------------------------------------------------------------------------
// Kernel 2: per-batch GEMM  S = Ee (T x V) * Pe^T (V x U)  via
// V_WMMA_F32_16X16X4_F32.  One wave computes a full 16x64 output strip
// (one t-tile x all 4 u-tiles): the A fragment is loaded once per K-step and
// reused for 4 WMMAs -> 1.25 loads/WMMA instead of 2, and exp-E is read from
// L2 once per strip instead of 4x.
//
// A (16x4 f32): lane l holds row M=l&15; VGPR0/1 = K = 2*(l>>4) + {0,1}.
// B (4x16 f32): lane l holds col N=l&15 (== row of Pe); same K split.
// => both fragments are contiguous float2 loads from row-major Ee / Pe.
// D (16x16 f32): VGPR r, lane l -> row M = r + 8*(l>>4), col N = l&15.
// ---------------------------------------------------------------------------
__global__ void rnnt_joint_wmma(const float* __restrict__ em, const float* __restrict__ pr,
                                const int* __restrict__ labels,
                                const float* __restrict__ Ee, const float* __restrict__ Pe,
                                const float* __restrict__ mE, const float* __restrict__ mP,
                                float* __restrict__ blank_lp, float* __restrict__ label_lp) {
    int tt = blockIdx.x;              // 0..7 : t-tile
    int b  = blockIdx.y;
    int t0 = tt * 16;
    int lane = threadIdx.x;
    int lo = lane & 15;
    int hi = lane >> 4;

    const float* Arow  = Ee + ((size_t)(b * T_ + t0 + lo)) * V_ + 2 * hi;
    const float* Brow0 = Pe + ((size_t)(b * U_ + lo)) * V_ + 2 * hi;

    v8f acc[4];
    #pragma unroll
    for (int ut = 0; ut < 4; ++ut) acc[ut] = (v8f)0.0f;

    #pragma unroll 4
    for (int k = 0; k < V_; k += 4) {
        v2f av = *(const v2f*)(Arow + k);
        #pragma unroll
        for (int ut = 0; ut < 4; ++ut) {
            v2f bv = *(const v2f*)(Brow0 + (size_t)ut * 16 * V_ + k);
            acc[ut] = __builtin_amdgcn_wmma_f32_16x16x4_f32(
                /*neg_a=*/false, av, /*neg_b=*/false, bv,
                /*c_mod=*/(short)0, acc[ut], /*reuse_a=*/false, /*reuse_b=*/false);
        }
    }

    // per-row (t) values shared across all 4 u-tiles: hoist out of ut loop
    float me8[8], e08[8];
    const float* erow8[8];
    #pragma unroll
    for (int r = 0; r < 8; ++r) {
        int t = t0 + r + 8 * hi;
        me8[r]   = mE[b * T_ + t];
        erow8[r] = em + ((size_t)(b * T_ + t)) * V_;
        e08[r]   = erow8[r][0];
    }

    #pragma unroll
    for (int ut = 0; ut < 4; ++ut) {
        int u = ut * 16 + lo;
        float n_u = mP[b * U_ + u];
        float p0  = pr[((size_t)(b * U_ + u)) * V_];      // predictions[b,u,0]
        bool haslbl = (u < U_ - 1);
        int lbl = 0;
        float plv = 0.0f;
        if (haslbl) {
            lbl = labels[b * (U_ - 1) + u];
            plv = pr[((size_t)(b * U_ + u)) * V_ + lbl];
        }
        #pragma unroll
        for (int r = 0; r < 8; ++r) {
            int t = t0 + r + 8 * hi;
            float lse = me8[r] + n_u + logf(acc[ut][r]);
            size_t oidx = ((size_t)(b * T_ + t)) * U_ + u;
            blank_lp[oidx] = e08[r] + p0 - lse;
            if (haslbl) label_lp[oidx] = erow8[r][lbl] + plv - lse;
        }
    }
}

// ---------------------------------------------------------------------------
// Kernel 3: forward recursion.  One 64-thread block per batch (thread = u).
// alpha[t,u] = c[u] + cumlogsumexp_{u'<=u}(a_top[u'] - c[u']),
// a_top = alpha[t-1,:]+blank[t-1,:],  c = exclusive cumsum of label_lp[t,:].
// Prefix scans via Hillis-Steele in LDS (6 rounds for U=64).
// ---------------------------------------------------------------------------
__device__ __forceinline__ float lae(float x, float y) {
    float mx = fmaxf(x, y);
    float mn = fminf(x, y);
    return mx + log1pf(expf(mn - mx));
}

__global__ void rnnt_forward(const float* __restrict__ blank_lp,
                             const float* __restrict__ label_lp,
                             const int* __restrict__ in_len,
                             const int* __restrict__ lab_len,
                             float* __restrict__ costs) {
    __shared__ float s[U_];
    int u = threadIdx.x;
    int b = blockIdx.x;
    int Tb = in_len[b];
    int Lb = lab_len[b];
    float a = (u == 0) ? 0.0f : NEG_INF;   // a_top carry

    for (int t = 0; t < T_; ++t) {
        size_t base = ((size_t)(b * T_ + t)) * U_;
        float blank_t = blank_lp[base + u];
        float lab = (u > 0) ? label_lp[base + u - 1] : 0.0f;

        // inclusive prefix-sum of shifted labels -> c[u] = sum_{k<u} label_t[k]
        s[u] = lab;
        __syncthreads();
        #pragma unroll
        for (int off = 1; off < U_; off <<= 1) {
            float add = (u >= off) ? s[u - off] : 0.0f;
            __syncthreads();
            s[u] += add;
            __syncthreads();
        }
        float c = s[u];

        // prefix log-sum-exp of (a - c)
        s[u] = a - c;
        __syncthreads();
        #pragma unroll
        for (int off = 1; off < U_; off <<= 1) {
            float add = (u >= off) ? s[u - off] : NEG_INF;
            __syncthreads();
            s[u] = lae(s[u], add);
            __syncthreads();
        }
        float alpha = c + s[u];
        a = alpha + blank_t;

        if (t == Tb - 1 && u == Lb) costs[b] = -a;
        __syncthreads();
    }
}

// ---------------------------------------------------------------------------
// Launch
// ---------------------------------------------------------------------------
extern "C" void kernel_launch(void* const* d_in, const int* in_sizes, int n_in,
                              void* d_out, int out_size, void* d_ws, size_t ws_size,
                              hipStream_t stream) {
    const float* em      = (const float*)d_in[0];   // (B,T,V)
    const float* pr      = (const float*)d_in[1];   // (B,U,V)
    const int*   labels  = (const int*)d_in[2];     // (B,U-1)
    const int*   in_len  = (const int*)d_in[3];     // (B,)
    const int*   lab_len = (const int*)d_in[4];     // (B,)

    float* ws = (float*)d_ws;
    float* Ee = ws;                                 // B*T*V
    float* Pe = Ee + (size_t)B_ * T_ * V_;          // B*U*V
    float* mE = Pe + (size_t)B_ * U_ * V_;          // B*T
    float* mP = mE + (size_t)B_ * T_;               // B*U
    float* blank_lp = mP + (size_t)B_ * U_;         // B*T*U
    float* label_lp = blank_lp + (size_t)B_ * T_ * U_; // B*T*U (u<U-1 used)

    rnnt_prep<<<B_ * T_ + B_ * U_, 128, 0, stream>>>(em, pr, Ee, Pe, mE, mP);

    dim3 g2(8, B_);
    rnnt_joint_wmma<<<g2, 32, 0, stream>>>(em, pr, labels, Ee, Pe, mE, mP,
                                           blank_lp, label_lp);

    rnnt_forward<<<B_, U_, 0, stream>>>(blank_lp, label_lp, in_len, lab_len,
                                        (float*)d_out);
}